// AI4DEM_22754736734808
// MI455X (gfx1250) — compile-verified
//
#include <hip/hip_runtime.h>
#include <stdint.h>

// Problem constants (match reference)
#define GN    2048
#define TILE  32
#define HALO  2
#define LTS   (TILE + 2*HALO)     // 36x36 LDS tile (with halo)

#define KN_F        100.0f
#define DT_F        1e-5f
#define DT_OVER_M   1e-3f         // DT / PARTICLE_MASS
#define EPLIS_F     1e-4f

// --- TDM availability probing -------------------------------------------------
#if defined(__has_builtin)
#  if __has_builtin(__builtin_amdgcn_tensor_load_to_lds)
#    define HAVE_TDM 1
#  endif
#endif
#ifndef HAVE_TDM
#  define HAVE_TDM 0
#endif

typedef unsigned int u32x4 __attribute__((ext_vector_type(4)));
typedef int          i32x4 __attribute__((ext_vector_type(4)));
typedef int          i32x8 __attribute__((ext_vector_type(8)));

// Build a 2D tensor D# and issue TENSOR_LOAD_TO_LDS for a LTSxLTS f32 tile.
__device__ __forceinline__ void tdm_load_tile_f32(uint32_t lds_byte_off,
                                                  const float* tile_base,
                                                  uint32_t row_stride_elems) {
#if defined(__HIP_DEVICE_COMPILE__) && HAVE_TDM
  uint64_t ga = (uint64_t)(uintptr_t)tile_base;

  // D# group 0: count=1 (valid user descriptor), lds_addr, global_addr, type=2
  u32x4 g0;
  g0[0] = 1u;                                            // count=1, no gather
  g0[1] = lds_byte_off;                                  // lds_addr (bytes)
  g0[2] = (uint32_t)ga;                                  // global_addr[31:0]
  g0[3] = ((uint32_t)(ga >> 32) & 0x01FFFFFFu)           // global_addr[56:32]
          | (2u << 30);                                  // type = 2 ("image")

  // D# group 1: data_size=4B; tensor dims large (no OOB); tile 36x36;
  // dim0 stride = array row stride.
  const uint32_t td0 = 1u << 20;
  const uint32_t td1 = 1u << 20;
  i32x8 g1;
  g1[0] = (int)(2u << 16);                               // data_size=2 -> 4 bytes
  g1[1] = (int)((td0 & 0xFFFFu) << 16);                  // tensor_dim0[15:0] @ [63:48]
  g1[2] = (int)((td0 >> 16) | ((td1 & 0xFFFFu) << 16));  // dim0 hi | dim1 lo
  g1[3] = (int)((td1 >> 16) | ((uint32_t)LTS << 16));    // dim1 hi | tile_dim0=36
  g1[4] = (int)LTS;                                      // tile_dim1=36, tile_dim2=0
  g1[5] = (int)row_stride_elems;                         // tensor_dim0_stride lo32
  g1[6] = 0;
  g1[7] = 0;

  i32x4 z4 = {0, 0, 0, 0};
#if __clang_major__ >= 23
  i32x8 z8 = {0, 0, 0, 0, 0, 0, 0, 0};
  __builtin_amdgcn_tensor_load_to_lds(g0, g1, z4, z4, z8, 0);
#else
  __builtin_amdgcn_tensor_load_to_lds(g0, g1, z4, z4, 0);
#endif
#else
  (void)lds_byte_off; (void)tile_base; (void)row_stride_elems;
#endif
}

__device__ __forceinline__ void wait_tensorcnt0() {
#if defined(__HIP_DEVICE_COMPILE__)
#  if defined(__has_builtin) && __has_builtin(__builtin_amdgcn_s_wait_tensorcnt)
  __builtin_amdgcn_s_wait_tensorcnt(0);
#  else
  asm volatile("s_wait_tensorcnt 0" ::: "memory");
#  endif
#endif
}

// Clamp a cell index into [-1, GN] so it fits i16 while preserving the
// "in [0,GN) or dropped" semantics of the scatter ops.
__device__ __forceinline__ int clamp_cell(int v) {
  v = v < -1 ? -1 : v;
  v = v > GN ? GN : v;
  return v;
}

// ---------------------------------------------------------------------------
// Kernel 0: zero the mover-list counter.
// ---------------------------------------------------------------------------
__global__ void dem_init_counter(unsigned* __restrict__ counter) {
  if (threadIdx.x == 0 && blockIdx.x == 0) *counter = 0u;
}

// ---------------------------------------------------------------------------
// Kernel 1: 5x5 contact-force stencil + integration (+ optional mover append).
// Writes: ws_x/ws_y = pristine updated positions (for the ordered scatter),
//         out_x/out_y/out_m initialized to x_new/y_new/mask,
//         list[] = packed {cy,cx,cyo,cxo} for every valid cell (if cap>0).
// ---------------------------------------------------------------------------
__global__ __launch_bounds__(256)
void dem_stencil_integrate(const float* __restrict__ x,
                           const float* __restrict__ y,
                           const float* __restrict__ vx,
                           const float* __restrict__ vy,
                           const float* __restrict__ mask,
                           float* __restrict__ ws_x,
                           float* __restrict__ ws_y,
                           float* __restrict__ out_x,
                           float* __restrict__ out_y,
                           float* __restrict__ out_m,
                           uint2* __restrict__ list,
                           unsigned* __restrict__ counter,
                           unsigned cap) {
  __shared__ float sx[LTS * LTS];
  __shared__ float sy[LTS * LTS];

  const int gx0 = blockIdx.x * TILE;
  const int gy0 = blockIdx.y * TILE;
  const int tx  = threadIdx.x & 31;        // column within tile
  const int ty0 = (threadIdx.x >> 5) * 4;  // 8 waves x 4 rows = 32 rows

  // Hoist the non-stencil operand loads so their latency hides under the TDM.
  float mk[4], vxv[4], vyv[4];
#pragma unroll
  for (int k = 0; k < 4; ++k) {
    const size_t g = (size_t)(gy0 + ty0 + k) * GN + (size_t)(gx0 + tx);
    mk[k]  = mask[g];
    vxv[k] = vx[g];
    vyv[k] = vy[g];
  }

  const bool interior = (blockIdx.x >= 1) && (blockIdx.x + 1 < gridDim.x) &&
                        (blockIdx.y >= 1) && (blockIdx.y + 1 < gridDim.y);

  bool tile_loaded = false;
#if defined(__HIP_DEVICE_COMPILE__) && HAVE_TDM
  if (interior) {
    if (threadIdx.x < 32) {   // wave 0 drives the TDM; EXEC is ignored by TDM ops
      const size_t base = (size_t)(gy0 - HALO) * GN + (size_t)(gx0 - HALO);
      tdm_load_tile_f32((uint32_t)(uintptr_t)&sx[0], x + base, GN);
      tdm_load_tile_f32((uint32_t)(uintptr_t)&sy[0], y + base, GN);
      wait_tensorcnt0();
    }
    tile_loaded = true;
  }
#endif
  if (!tile_loaded) {
    // Edge tiles (periodic wrap) / no-TDM fallback: cooperative LDS fill.
    for (int idx = threadIdx.x; idx < LTS * LTS; idx += 256) {
      const int r  = idx / LTS;
      const int c  = idx - r * LTS;
      const int gr = (gy0 - HALO + r) & (GN - 1);
      const int gc = (gx0 - HALO + c) & (GN - 1);
      const size_t g = (size_t)gr * GN + gc;
      sx[idx] = x[g];
      sy[idx] = y[g];
    }
  }
  __syncthreads();

#pragma unroll
  for (int k = 0; k < 4; ++k) {
    const int r = ty0 + k;
    const int c = tx;
    const float xc = sx[(r + HALO) * LTS + (c + HALO)];
    const float yc = sy[(r + HALO) * LTS + (c + HALO)];

    float fx = 0.0f, fy = 0.0f;
#pragma unroll
    for (int a = -2; a <= 2; ++a) {
#pragma unroll
      for (int b = -2; b <= 2; ++b) {
        const float xn = sx[(r + HALO + a) * LTS + (c + HALO + b)];
        const float yn = sy[(r + HALO + a) * LTS + (c + HALO + b)];
        const float dx = xc - xn;
        const float dy = yc - yn;
        const float dist = sqrtf(dx * dx + dy * dy);
        const float mag  = __fdividef(KN_F * (dist - 2.0f), fmaxf(EPLIS_F, dist));
        const bool contact = dist < 2.0f;
        fx += contact ? mag * dx : 0.0f;
        fy += contact ? mag * dy : 0.0f;
      }
    }

    const size_t g = (size_t)(gy0 + r) * GN + (size_t)(gx0 + c);
    const float nvx = vxv[k] - DT_OVER_M * fx * mk[k];
    const float nvy = vyv[k] - DT_OVER_M * fy * mk[k];
    const float xnw = xc + DT_F * nvx;
    const float ynw = yc + DT_F * nvy;

    ws_x[g]  = xnw;     // pristine copy for the ordered gather
    ws_y[g]  = ynw;
    out_x[g] = xnw;     // xg/yg/m initial state before the scatter ops
    out_y[g] = ynw;
    out_m[g] = mk[k];

    if (cap != 0u) {
      const int cxo = (int)xc;   // truncation toward zero == astype(int32)
      const int cyo = (int)yc;
      const int cx  = (int)xnw;
      const int cy  = (int)ynw;
      if (cx != 0 && cy != 0 && cxo != 0 && cyo != 0) {
        const unsigned slot = atomicAdd(counter, 1u);
        if (slot < cap) {
          uint2 e;
          e.x = ((unsigned)(clamp_cell(cy)  & 0xFFFF) << 16) | (unsigned)(clamp_cell(cx)  & 0xFFFF);
          e.y = ((unsigned)(clamp_cell(cyo) & 0xFFFF) << 16) | (unsigned)(clamp_cell(cxo) & 0xFFFF);
          list[slot] = e;
        }
      }
    }
  }
}

// ---------------------------------------------------------------------------
// List-driven scatter pass 2: m[cy,cx]=1 ; gather ws[cyo,cxo] ; write xg/yg.
// ---------------------------------------------------------------------------
__global__ __launch_bounds__(256)
void dem_scatter_new_list(const uint2* __restrict__ list,
                          const unsigned* __restrict__ counter,
                          unsigned cap,
                          const float* __restrict__ ws_x,
                          const float* __restrict__ ws_y,
                          float* __restrict__ out_x,
                          float* __restrict__ out_y,
                          float* __restrict__ out_m) {
  const unsigned n = min(*counter, cap);
  const unsigned i = blockIdx.x * blockDim.x + threadIdx.x;
  if (i >= n) return;

  const uint2 e = list[i];
  const int cx  = (short)(e.x & 0xFFFFu);
  const int cy  = (short)(e.x >> 16);
  const int cxo = (short)(e.y & 0xFFFFu);
  const int cyo = (short)(e.y >> 16);

  float gx = 0.0f, gy = 0.0f;                              // mode='fill', 0.0
  if ((unsigned)cxo < GN && (unsigned)cyo < GN) {
    const size_t s = (size_t)cyo * GN + (size_t)cxo;
    gx = ws_x[s];
    gy = ws_y[s];
  }
  if ((unsigned)cx < GN && (unsigned)cy < GN) {            // mode='drop'
    const size_t t = (size_t)cy * GN + (size_t)cx;
    out_m[t] = 1.0f;
    out_x[t] = gx;
    out_y[t] = gy;
  }
}

// ---------------------------------------------------------------------------
// List-driven scatter pass 3: clear old cell (strictly after pass 2).
// ---------------------------------------------------------------------------
__global__ __launch_bounds__(256)
void dem_clear_old_list(const uint2* __restrict__ list,
                        const unsigned* __restrict__ counter,
                        unsigned cap,
                        float* __restrict__ out_x,
                        float* __restrict__ out_y,
                        float* __restrict__ out_m) {
  const unsigned n = min(*counter, cap);
  const unsigned i = blockIdx.x * blockDim.x + threadIdx.x;
  if (i >= n) return;

  const uint2 e = list[i];
  const int cxo = (short)(e.y & 0xFFFFu);
  const int cyo = (short)(e.y >> 16);

  if ((unsigned)cxo < GN && (unsigned)cyo < GN) {          // mode='drop'
    const size_t s = (size_t)cyo * GN + (size_t)cxo;
    out_m[s] = 0.0f;
    out_x[s] = 0.0f;
    out_y[s] = 0.0f;
  }
}

// ---------------------------------------------------------------------------
// Fallback full-grid scatter passes (used if ws is too small for the list).
// ---------------------------------------------------------------------------
__global__ __launch_bounds__(256)
void dem_scatter_new(const float* __restrict__ x_old,
                     const float* __restrict__ y_old,
                     const float* __restrict__ ws_x,
                     const float* __restrict__ ws_y,
                     float* __restrict__ out_x,
                     float* __restrict__ out_y,
                     float* __restrict__ out_m) {
  const int i = blockIdx.x * blockDim.x + threadIdx.x;
  if (i >= GN * GN) return;

  const int cxo = (int)x_old[i];
  const int cyo = (int)y_old[i];
  const int cx  = (int)ws_x[i];
  const int cy  = (int)ws_y[i];

  if (cx == 0 || cy == 0 || cxo == 0 || cyo == 0) return;

  float gx = 0.0f, gy = 0.0f;
  if ((unsigned)cxo < GN && (unsigned)cyo < GN) {
    const size_t s = (size_t)cyo * GN + (size_t)cxo;
    gx = ws_x[s];
    gy = ws_y[s];
  }
  if ((unsigned)cx < GN && (unsigned)cy < GN) {
    const size_t t = (size_t)cy * GN + (size_t)cx;
    out_m[t] = 1.0f;
    out_x[t] = gx;
    out_y[t] = gy;
  }
}

__global__ __launch_bounds__(256)
void dem_clear_old(const float* __restrict__ x_old,
                   const float* __restrict__ y_old,
                   const float* __restrict__ ws_x,
                   const float* __restrict__ ws_y,
                   float* __restrict__ out_x,
                   float* __restrict__ out_y,
                   float* __restrict__ out_m) {
  const int i = blockIdx.x * blockDim.x + threadIdx.x;
  if (i >= GN * GN) return;

  const int cxo = (int)x_old[i];
  const int cyo = (int)y_old[i];
  const int cx  = (int)ws_x[i];
  const int cy  = (int)ws_y[i];

  if (cx == 0 || cy == 0 || cxo == 0 || cyo == 0) return;

  if ((unsigned)cxo < GN && (unsigned)cyo < GN) {
    const size_t s = (size_t)cyo * GN + (size_t)cxo;
    out_m[s] = 0.0f;
    out_x[s] = 0.0f;
    out_y[s] = 0.0f;
  }
}

// ---------------------------------------------------------------------------
extern "C" void kernel_launch(void* const* d_in, const int* in_sizes, int n_in,
                              void* d_out, int out_size, void* d_ws, size_t ws_size,
                              hipStream_t stream) {
  (void)in_sizes; (void)n_in; (void)out_size;

  const float* x_in  = (const float*)d_in[0];
  const float* y_in  = (const float*)d_in[1];
  const float* vx_in = (const float*)d_in[2];
  const float* vy_in = (const float*)d_in[3];
  const float* mask  = (const float*)d_in[4];

  const size_t nn = (size_t)GN * GN;
  float* out_x = (float*)d_out;                 // xg
  float* out_y = out_x + nn;                    // yg
  float* out_m = out_y + nn;                    // m

  // Workspace layout: ws_x | ws_y | (128B-aligned) counter | list
  float* ws_x = (float*)d_ws;
  float* ws_y = ws_x + nn;
  uintptr_t pa = ((uintptr_t)(ws_y + nn) + 127u) & ~(uintptr_t)127u;
  unsigned* counter = (unsigned*)pa;
  uint2*    list    = (uint2*)(pa + 128u);

  const uintptr_t used = (pa + 128u) - (uintptr_t)d_ws;
  size_t avail = (ws_size > used) ? (size_t)(ws_size - used) : 0;
  size_t cap_sz = avail / sizeof(uint2);
  // Enable the compacted path only if the worst case (every cell valid) fits.
  unsigned cap = (cap_sz >= nn) ? (unsigned)nn : 0u;

  dim3 grid(GN / TILE, GN / TILE);
  dim3 block(256);
  const int b1 = 256;
  const int g1 = (int)((nn + b1 - 1) / b1);

  if (cap) dem_init_counter<<<1, 1, 0, stream>>>(counter);

  dem_stencil_integrate<<<grid, block, 0, stream>>>(x_in, y_in, vx_in, vy_in, mask,
                                                    ws_x, ws_y, out_x, out_y, out_m,
                                                    list, counter, cap);

  if (cap) {
    dem_scatter_new_list<<<g1, b1, 0, stream>>>(list, counter, cap, ws_x, ws_y,
                                                out_x, out_y, out_m);
    dem_clear_old_list <<<g1, b1, 0, stream>>>(list, counter, cap,
                                               out_x, out_y, out_m);
  } else {
    dem_scatter_new<<<g1, b1, 0, stream>>>(x_in, y_in, ws_x, ws_y, out_x, out_y, out_m);
    dem_clear_old <<<g1, b1, 0, stream>>>(x_in, y_in, ws_x, ws_y, out_x, out_y, out_m);
  }
}